// Model_50225347559738
// MI455X (gfx1250) — compile-verified
//
#include <hip/hip_runtime.h>
#include <hip/hip_bf16.h>

typedef __attribute__((ext_vector_type(2))) float v2f;
typedef __attribute__((ext_vector_type(8))) float v8f;

#define NITEMS    30000
#define KNBR      32
#define DIM       64
#define NRELS     40
#define UVSTRIDE  132           // padded LDS row stride (floats) to avoid bank conflicts
#define NEG_SLOPE 0.2f
#define MASK_VAL  -9000000000000000.0f

// ---------------------------------------------------------------------------
// Kernel 1: relation-projection precompute with V_WMMA_F32_16X16X4_F32.
//   UV (40x128) = relEmbs (40x64) @ W_w (64x128)     [u_j | v_j]
//   c  (40)     = relEmbs @ W_b
// Grid: (8 n-tiles, 3 m-tiles), 32 threads (one wave) per block.
// Branch-free: OOB rows load from a clamped address and are zeroed by select.
// ---------------------------------------------------------------------------
__global__ __launch_bounds__(32) void relproj_wmma(
    const float* __restrict__ relEmbs,   // (40,64)
    const float* __restrict__ W_w,       // (64,128)
    const float* __restrict__ W_b,       // (64)
    float* __restrict__ UV,              // (40,128) out
    float* __restrict__ cvec)            // (40) out
{
    const int lane = threadIdx.x;            // 0..31
    const int tn = blockIdx.x;               // 0..7  (N tile of 128)
    const int tm = blockIdx.y;               // 0..2  (M tile of 48 >= 40)
    const int m15  = lane & 15;
    const int mrow = tm * 16 + m15;          // A-matrix row (relation id)
    const int ncol = tn * 16 + m15;          // B/C column
    const int khalf = (lane >> 4) << 1;      // 0 for lanes 0-15, 2 for lanes 16-31

    const bool mvalid = (mrow < NRELS);
    const int  mrowc  = mvalid ? mrow : (NRELS - 1);   // clamped, always in-bounds
    const float* __restrict__ arow = relEmbs + mrowc * DIM;

    v8f acc = {};
#pragma unroll
    for (int kk = 0; kk < 16; ++kk) {        // K = 64 in steps of 4
        const int k0 = kk * 4 + khalf;
        // A fragment (16x4 f32): lanes 0-15 hold K0,K1; lanes 16-31 hold K2,K3.
        // Unconditional b64 load from clamped row, then select-zero (no branch).
        const float2 ap = *(const float2*)(arow + k0);
        v2f a, b;
        a.x = mvalid ? ap.x : 0.0f;
        a.y = mvalid ? ap.y : 0.0f;
        // B fragment (4x16 f32): same K split across lane halves, N = lane&15
        b.x = W_w[(k0    ) * 128 + ncol];
        b.y = W_w[(k0 + 1) * 128 + ncol];
        acc = __builtin_amdgcn_wmma_f32_16x16x4_f32(
            /*neg_a=*/false, a, /*neg_b=*/false, b,
            /*c_mod=*/(short)0, acc, /*reuse_a=*/false, /*reuse_b=*/false);
    }

    // C/D layout: VGPR r -> M = r (lanes 0-15) / r+8 (lanes 16-31), N = lane&15
    const int rbase = tm * 16 + ((lane >> 4) ? 8 : 0);
#pragma unroll
    for (int r = 0; r < 8; ++r) {
        const int row = rbase + r;
        if (row < NRELS) UV[row * 128 + ncol] = acc[r];
    }

    // c_j = relEmbs[j] . W_b  (tiny; one block does it)
    if (tm == 0 && tn == 0) {
        for (int j = lane; j < NRELS; j += 32) {
            float s = 0.0f;
            for (int e = 0; e < DIM; ++e) s += relEmbs[j * DIM + e] * W_b[e];
            cvec[j] = s;
        }
    }
}

// ---------------------------------------------------------------------------
// Kernel 2: gather + dot + masked wave32 softmax.
// One wave per item (lane = neighbor k). 8 items per 256-thread block.
// All gathers are unconditional from clamped addresses; masking via selects.
// ---------------------------------------------------------------------------
__global__ __launch_bounds__(256) void attn_kernel(
    const float* __restrict__ entiEmbs,       // (80000,64)
    const int*   __restrict__ item_ids,       // (30000)
    const int*   __restrict__ item_entities,  // (30000,32)
    const int*   __restrict__ item_relations, // (30000,32)
    const int*   __restrict__ n_entities_p,   // scalar
    const float* __restrict__ UV,             // (40,128) in ws
    const float* __restrict__ cvec,           // (40) in ws
    float*       __restrict__ out)            // (30000,32)
{
    __shared__ float sUV[NRELS * UVSTRIDE];   // 21120 B, padded rows
    __shared__ float sC[NRELS];
    __shared__ float sH[8][DIM];              // one h row per wave

    const int tid  = threadIdx.x;
    const int wave = tid >> 5;
    const int lane = tid & 31;
    const int item = blockIdx.x * 8 + wave;   // 3750*8 == 30000 exactly
    const int nent = n_entities_p[0];

    // Stage UV table into LDS with padded stride
    for (int i = tid; i < NRELS * 128; i += 256) {
        const int r = i >> 7, c = i & 127;
        sUV[r * UVSTRIDE + c] = UV[i];
    }
    if (tid < NRELS) sC[tid] = cvec[tid];

    // Stage item embedding h (64 floats) per wave: each lane writes 2 floats.
    // Unconditional load from clamped row, masked by select.
    {
        const int  hid    = item_ids[item];
        const bool hvalid = (hid >= 0 && hid < nent);
        const int  hidc   = hvalid ? hid : 0;
        const float2 hp = *(const float2*)(entiEmbs + (long long)hidc * DIM + lane * 2);
        sH[wave][lane * 2]     = hvalid ? hp.x : 0.0f;
        sH[wave][lane * 2 + 1] = hvalid ? hp.y : 0.0f;
    }
    __syncthreads();

    const int  idx    = item * KNBR + lane;
    const int  ent    = item_entities[idx];
    int        rel    = item_relations[idx] - 1;
    rel = min(max(rel, 0), NRELS - 1);
    const bool valid  = (ent != nent);
    const bool tvalid = (ent >= 0 && ent < nent);   // pad row (== nent) is virtual zeros
    const int  entc   = tvalid ? ent : 0;           // clamped, always in-bounds

    const float4* __restrict__ trow = (const float4*)(entiEmbs + (long long)entc * DIM);
    const float4* __restrict__ up   = (const float4*)(&sUV[rel * UVSTRIDE]);       // u_j
    const float4* __restrict__ vp   = (const float4*)(&sUV[rel * UVSTRIDE + 64]);  // v_j
    const float4* __restrict__ hp   = (const float4*)(sH[wave]);

    float hs = 0.0f;   // u . h
    float ts = 0.0f;   // v . t  (masked at the end)
#pragma unroll
    for (int d4 = 0; d4 < 16; ++d4) {
        const float4 tv = trow[d4];     // unconditional global b128
        const float4 hv = hp[d4];       // LDS b128 (wave-uniform row)
        const float4 uu = up[d4];       // LDS b128
        const float4 vv = vp[d4];       // LDS b128
        hs += hv.x * uu.x + hv.y * uu.y + hv.z * uu.z + hv.w * uu.w;
        ts += tv.x * vv.x + tv.y * vv.y + tv.z * vv.z + tv.w * vv.w;
    }
    const float s = sC[rel] + hs + (tvalid ? ts : 0.0f);

    // leaky ReLU + mask
    const float e = (s > 0.0f) ? s : NEG_SLOPE * s;
    const float x = valid ? e : MASK_VAL;

    // wave32 softmax over the 32 lanes (one item per wave)
    float m = x;
#pragma unroll
    for (int off = 16; off > 0; off >>= 1) m = fmaxf(m, __shfl_xor(m, off, 32));
    const float ex = __expf(x - m);
    float sum = ex;
#pragma unroll
    for (int off = 16; off > 0; off >>= 1) sum += __shfl_xor(sum, off, 32);

    out[idx] = ex / sum;
}

// ---------------------------------------------------------------------------
extern "C" void kernel_launch(void* const* d_in, const int* in_sizes, int n_in,
                              void* d_out, int out_size, void* d_ws, size_t ws_size,
                              hipStream_t stream) {
    const float* entiEmbs       = (const float*)d_in[0];
    const float* relEmbs        = (const float*)d_in[1];
    const float* W_w            = (const float*)d_in[2];
    const float* W_b            = (const float*)d_in[3];
    const int*   item_ids       = (const int*)d_in[4];
    const int*   item_entities  = (const int*)d_in[5];
    const int*   item_relations = (const int*)d_in[6];
    const int*   n_entities     = (const int*)d_in[7];
    float* out = (float*)d_out;

    float* UV   = (float*)d_ws;          // 40*128 floats
    float* cvec = UV + NRELS * 128;      // 40 floats  (total ~20.6 KB of ws)

    dim3 g1(8, 3);
    relproj_wmma<<<g1, 32, 0, stream>>>(relEmbs, W_w, W_b, UV, cvec);

    attn_kernel<<<NITEMS / 8, 256, 0, stream>>>(entiEmbs, item_ids, item_entities,
                                                item_relations, n_entities,
                                                UV, cvec, out);
}